// GCN_36000415875141
// MI455X (gfx1250) — compile-verified
//
#include <hip/hip_runtime.h>
#include <hip/hip_bf16.h>

// ---------------------------------------------------------------------------
// GCN 3-layer forward for MI455X (gfx1250, wave32).
//   layer: h = x @ W   (WMMA f32 16x16x4, fp32-exact, B register-resident)
//          agg = sum_{e:dst=i} dinv[src]*dinv[dst]*h[src] + dinv[i]^2*h[i]
//          y = relu(bn(agg + b))          (layers 1,2;  layer 3: agg + b)
// Aggregation is L2-resident (25.6 MB features << 192 MB L2) -> f32 atomics.
// ---------------------------------------------------------------------------

typedef __attribute__((ext_vector_type(2))) float v2f;
typedef __attribute__((ext_vector_type(8))) float v8f;

static constexpr int GN  = 50000;   // nodes
static constexpr int GE  = 800000;  // edges
static constexpr int GIN = 128;
static constexpr int GH  = 128;
static constexpr int GC  = 64;

// ---------------- degree / dinv ----------------

__global__ void deg_init_kernel(float* __restrict__ deg, int n) {
    int i = blockIdx.x * blockDim.x + threadIdx.x;
    if (i < n) deg[i] = 1.0f;                      // self-loop contributes 1
}

__global__ void deg_count_kernel(const int* __restrict__ dst, float* __restrict__ deg, int nE) {
    int e = blockIdx.x * blockDim.x + threadIdx.x;
    if (e < nE) atomicAdd(&deg[dst[e]], 1.0f);
}

__global__ void deg_to_dinv_kernel(float* __restrict__ deg, int n) {
    int i = blockIdx.x * blockDim.x + threadIdx.x;
    if (i < n) deg[i] = rsqrtf(deg[i]);            // deg >= 1 always
}

// ---------------- persistent WMMA GEMM: Y[n,DOUT] = X[n,DIN] @ W[DIN,DOUT] --
// Each wave owns one 16-wide N-column (tn) and keeps ALL B fragments for that
// column register-resident (DIN/4 x v2f = 64 VGPRs for DIN=128), then
// grid-strides over M-tiles.  Inner loop: 1 global_load_b64 (A) per WMMA.

template <int DIN, int DOUT>
__global__ void gemm_wmma_f32_kernel(const float* __restrict__ X,
                                     const float* __restrict__ W,
                                     float* __restrict__ Y,
                                     int mTiles) {
    constexpr int tilesN = DOUT / 16;
    constexpr int KF     = DIN / 4;      // number of K=4 fragments

    const int wave       = threadIdx.x >> 5;
    const int wavesPerBl = blockDim.x >> 5;
    const int wid        = blockIdx.x * wavesPerBl + wave;
    const int wavesTotal = gridDim.x * wavesPerBl;
    const int tn         = wid % tilesN;
    const int mStep      = wavesTotal / tilesN;     // waves sharing a column

    const int lane = threadIdx.x & 31;
    const int half = lane >> 4;          // 0: lanes 0-15, 1: lanes 16-31
    const int l15  = lane & 15;
    const int col  = tn * 16 + l15;      // B/C/D N index for this lane

    // ---- hoist B: load all fragments for this column once ----
    v2f bfrag[KF];
#pragma unroll
    for (int j = 0; j < KF; ++j) {
        const int k = 4 * j + 2 * half;  // VGPR0=K(+2*half), VGPR1=K+1(+2*half)
        bfrag[j].x = W[(size_t)k * DOUT + col];
        bfrag[j].y = W[(size_t)(k + 1) * DOUT + col];
    }

    // ---- persistent loop over M-tiles ----
    for (int tm = wid / tilesN; tm < mTiles; tm += mStep) {
        const float* __restrict__ xr =
            X + (size_t)(tm * 16 + l15) * DIN + 2 * half;
        v8f acc = {};
#pragma unroll
        for (int j = 0; j < KF; ++j) {
            v2f a;                       // contiguous 8B load per lane
            a.x = xr[4 * j];
            a.y = xr[4 * j + 1];
            acc = __builtin_amdgcn_wmma_f32_16x16x4_f32(
                /*neg_a=*/false, a, /*neg_b=*/false, bfrag[j],
                /*c_mod=*/(short)0, acc, /*reuse_a=*/false, /*reuse_b=*/false);
        }
        // C/D layout: VGPR r holds M = r + 8*half, N = lane&15
        float* __restrict__ yr = Y + (size_t)(tm * 16 + 8 * half) * DOUT + col;
#pragma unroll
        for (int r = 0; r < 8; ++r)
            yr[(size_t)r * DOUT] = acc[r];
    }
}

// ---------------- self-loop init (dense, no atomics) ----------------
// agg[i][f] = dinv[i]^2 * h[i][f]  (+ bias[f] if ADD_BIAS)

template <int D, bool ADD_BIAS>
__global__ void selfloop_init_kernel(const float* __restrict__ h,
                                     const float* __restrict__ dinv,
                                     const float* __restrict__ bias,
                                     float* __restrict__ agg, int n) {
    size_t idx = (size_t)blockIdx.x * blockDim.x + threadIdx.x; // one float4
    size_t total = (size_t)n * D / 4;
    if (idx >= total) return;
    size_t base = idx * 4;
    int i = (int)(base / D);
    int f = (int)(base % D);
    float di = dinv[i];
    float w  = di * di;
    float4 v = *(const float4*)(h + base);
    float4 o;
    o.x = w * v.x; o.y = w * v.y; o.z = w * v.z; o.w = w * v.w;
    if (ADD_BIAS) {
        o.x += bias[f]; o.y += bias[f + 1]; o.z += bias[f + 2]; o.w += bias[f + 3];
    }
    *(float4*)(agg + base) = o;
}

// ---------------- per-edge scatter: one wave per edge ----------------
// lane owns D/32 consecutive features; float4/float2 gather + f32 atomics.

template <int D>
__global__ void edge_scatter_kernel(const int* __restrict__ src,
                                    const int* __restrict__ dst,
                                    const float* __restrict__ dinv,
                                    const float* __restrict__ h,
                                    float* __restrict__ agg, int nE) {
    constexpr int VPT = D / 32;
    static_assert(VPT == 2 || VPT == 4, "D must be 64 or 128");
    const int wave = threadIdx.x >> 5;
    const int lane = threadIdx.x & 31;
    const int e    = blockIdx.x * (blockDim.x >> 5) + wave;
    if (e >= nE) return;

    const int s = src[e];
    const int d = dst[e];
    const float norm = dinv[s] * dinv[d];

    const float* __restrict__ hp = h   + (size_t)s * D + lane * VPT;
    float* __restrict__ ap       = agg + (size_t)d * D + lane * VPT;

    if (VPT == 4) {
        float4 v = *(const float4*)hp;
        atomicAdd(ap + 0, norm * v.x);
        atomicAdd(ap + 1, norm * v.y);
        atomicAdd(ap + 2, norm * v.z);
        atomicAdd(ap + 3, norm * v.w);
    } else {
        float2 v = *(const float2*)hp;
        atomicAdd(ap + 0, norm * v.x);
        atomicAdd(ap + 1, norm * v.y);
    }
}

// ---------------- fused bias + BN(eval) + ReLU ----------------

template <int D>
__global__ void bias_bn_relu_kernel(float* __restrict__ agg,
                                    const float* __restrict__ b,
                                    const float* __restrict__ g,
                                    const float* __restrict__ be,
                                    const float* __restrict__ m,
                                    const float* __restrict__ v, int n) {
    size_t idx = (size_t)blockIdx.x * blockDim.x + threadIdx.x; // one float4
    size_t total = (size_t)n * D / 4;
    if (idx >= total) return;
    size_t base = idx * 4;
    int f = (int)(base % D);
    float4 a = *(float4*)(agg + base);
    float* pa = &a.x;
#pragma unroll
    for (int j = 0; j < 4; ++j) {
        int fj = f + j;
        float val = pa[j] + b[fj];
        val = g[fj] * (val - m[fj]) * rsqrtf(v[fj] + 1e-5f) + be[fj];
        pa[j] = fmaxf(val, 0.0f);
    }
    *(float4*)(agg + base) = a;
}

// ---------------------------------------------------------------------------

static inline size_t align256(size_t x) { return (x + 255) & ~(size_t)255; }

extern "C" void kernel_launch(void* const* d_in, const int* in_sizes, int n_in,
                              void* d_out, int out_size, void* d_ws, size_t ws_size,
                              hipStream_t stream) {
    (void)in_sizes; (void)n_in; (void)out_size; (void)ws_size;

    const float* x   = (const float*)d_in[0];
    const int*   ei  = (const int*)d_in[1];
    const int*   src = ei;          // edge_index[0]
    const int*   dst = ei + GE;     // edge_index[1]
    const float* W1  = (const float*)d_in[2];
    const float* b1  = (const float*)d_in[3];
    const float* g1  = (const float*)d_in[4];
    const float* be1 = (const float*)d_in[5];
    const float* m1  = (const float*)d_in[6];
    const float* v1  = (const float*)d_in[7];
    const float* W2  = (const float*)d_in[8];
    const float* b2  = (const float*)d_in[9];
    const float* g2  = (const float*)d_in[10];
    const float* be2 = (const float*)d_in[11];
    const float* m2  = (const float*)d_in[12];
    const float* v2  = (const float*)d_in[13];
    const float* W3  = (const float*)d_in[14];
    const float* b3  = (const float*)d_in[15];
    float* out = (float*)d_out;

    // workspace carve-up (all 256B aligned): dinv | hbuf | agg1 | agg2
    char* ws = (char*)d_ws;
    size_t off = 0;
    float* dinv = (float*)(ws + off); off += align256((size_t)GN * 4);
    float* hbuf = (float*)(ws + off); off += align256((size_t)GN * GH * 4);
    float* agg1 = (float*)(ws + off); off += align256((size_t)GN * GH * 4);
    float* agg2 = (float*)(ws + off); off += align256((size_t)GN * GH * 4);

    const int TPB = 256;

    // ---- degree -> dinv (shared by all layers) ----
    deg_init_kernel<<<(GN + TPB - 1) / TPB, TPB, 0, stream>>>(dinv, GN);
    deg_count_kernel<<<(GE + TPB - 1) / TPB, TPB, 0, stream>>>(dst, dinv, GE);
    deg_to_dinv_kernel<<<(GN + TPB - 1) / TPB, TPB, 0, stream>>>(dinv, GN);

    const int mTiles = GN / 16;                  // 3125 (exact)
    const int gemmBlocks = 512;                  // 4096 persistent waves
    const int edgeBlocks = (GE + 7) / 8;         // 8 waves/block, 1 edge/wave
    const int elemBlocksH = (GN * GH / 4 + TPB - 1) / TPB;
    const int elemBlocksC = (GN * GC / 4 + TPB - 1) / TPB;

    // ---- layer 1: x @ W1, aggregate, bias+BN+ReLU ----
    gemm_wmma_f32_kernel<GIN, GH><<<gemmBlocks, TPB, 0, stream>>>(
        x, W1, hbuf, mTiles);
    selfloop_init_kernel<GH, false><<<elemBlocksH, TPB, 0, stream>>>(
        hbuf, dinv, nullptr, agg1, GN);
    edge_scatter_kernel<GH><<<edgeBlocks, TPB, 0, stream>>>(
        src, dst, dinv, hbuf, agg1, GE);
    bias_bn_relu_kernel<GH><<<elemBlocksH, TPB, 0, stream>>>(
        agg1, b1, g1, be1, m1, v1, GN);

    // ---- layer 2: y1 @ W2, aggregate, bias+BN+ReLU ----
    gemm_wmma_f32_kernel<GH, GH><<<gemmBlocks, TPB, 0, stream>>>(
        agg1, W2, hbuf, mTiles);
    selfloop_init_kernel<GH, false><<<elemBlocksH, TPB, 0, stream>>>(
        hbuf, dinv, nullptr, agg2, GN);
    edge_scatter_kernel<GH><<<edgeBlocks, TPB, 0, stream>>>(
        src, dst, dinv, hbuf, agg2, GE);
    bias_bn_relu_kernel<GH><<<elemBlocksH, TPB, 0, stream>>>(
        agg2, b2, g2, be2, m2, v2, GN);

    // ---- layer 3: y2 @ W3, aggregate (+bias folded into init) -> d_out ----
    gemm_wmma_f32_kernel<GH, GC><<<gemmBlocks, TPB, 0, stream>>>(
        agg2, W3, hbuf, mTiles);
    selfloop_init_kernel<GC, true><<<elemBlocksC, TPB, 0, stream>>>(
        hbuf, dinv, b3, out, GN);
    edge_scatter_kernel<GC><<<edgeBlocks, TPB, 0, stream>>>(
        src, dst, dinv, hbuf, out, GE);
}